// SRP_36618891166068
// MI455X (gfx1250) — compile-verified
//
#include <hip/hip_runtime.h>
#include <math.h>

typedef float v2f __attribute__((ext_vector_type(2)));
typedef float v8f __attribute__((ext_vector_type(8)));

#define NSAMP  256000
#define NCH    8
#define NFFT   512
#define HOP    256
#define NT     1001     // STFT frames
#define NTP    1008     // padded frames (63*16)
#define NTT    63       // t tiles of 16
#define NF     256      // freq bins used (rfft bins 1..256, DC dropped)
#define NA     361      // angles
#define NAP    368      // padded angles (23*16)
#define NAT    23       // angle tiles of 16
#define FROW   516      // padded LDS frame row (floats) -> bank-conflict free
#define TWO_PI 6.28318530717958647692f

// D = A(16x4) * B(4x16) + C, fp32, wave32.
// ISA layouts: A: M=lane%16, K=2*(lane/16)+v ; B: N=lane%16, K=2*(lane/16)+v ;
//              C/D: N=lane%16, M=8*(lane/16)+r
__device__ __forceinline__ v8f wmma_f32(v2f a, v2f b, v8f c) {
  // f32 WMMA: NEG for A/B must be 0 (only CNeg allowed) -> pass false.
  return __builtin_amdgcn_wmma_f32_16x16x4_f32(false, a, false, b, (short)0, c,
                                               false, false);
}

// ---------------------------------------------------------------- DFT matrix
// K-pair-interleaved layouts so a B fragment is ONE aligned b64 load:
//   Bc[p][fi][2] = { cos(2pi*2p*(fi+1)/512),  cos(2pi*(2p+1)*(fi+1)/512) }
//   Bs[p][fi][2] = { -sin(...2p...),          -sin(...2p+1...) }
__global__ void k_dft_init(float* __restrict__ Bc, float* __restrict__ Bs) {
  int n  = blockIdx.x;    // 0..511
  int fi = threadIdx.x;   // 0..255
  int k  = (n * (fi + 1)) & (NFFT - 1);   // exact integer argument reduction
  float ang = (float)k * (TWO_PI / (float)NFFT);
  float s, c;
  sincosf(ang, &s, &c);
  size_t idx = (((size_t)(n >> 1) * NF) + fi) * 2 + (n & 1);
  Bc[idx] = c;
  Bs[idx] = -s;
}

// ------------------------------------------------------------ steering init
// Stacked-complex K=16 rows per freq (row k<8: channel k; row k>=8: ch k-8):
//   Bre rows: [cos(phi_c) | sin(phi_c)] ;  Bim rows: [-sin(phi_c) | cos(phi_c)]
// Stored pair-interleaved: B[f][p][a][2] = {row 2p, row 2p+1}, p=0..7.
__global__ void __launch_bounds__(384) k_steer_init(float* __restrict__ Bre,
                                                    float* __restrict__ Bim) {
  int f = blockIdx.x;     // 0..255  (analog freq = (f+1)*31.25 Hz)
  int a = threadIdx.x;    // 0..383
  if (a >= NAP) return;
  bool valid = (a < NA);
  float st = 0.0f;
  if (valid) {
    float theta = (-90.0f + 0.5f * (float)a) * 0.01745329251994329577f;
    st = sinf(theta);
  }
  float fa = (float)(f + 1) * 31.25f;
  float K0 = TWO_PI * (0.04f / 343.0f) * fa;   // 2*pi*d/c * f
  auto put = [&](float* base, int k, float v) {
    base[(((size_t)f * 8 + (k >> 1)) * NAP + a) * 2 + (k & 1)] = v;
  };
#pragma unroll
  for (int c = 0; c < 8; ++c) {
    float sphi = 0.0f, cphi = 0.0f;
    if (valid) {
      float phi = K0 * st * (float)c;
      sincosf(phi, &sphi, &cphi);
    }
    put(Bre, c, cphi);
    put(Bre, 8 + c, sphi);
    put(Bim, c, -sphi);
    put(Bim, 8 + c, cphi);
  }
}

// ----------------------------------------------------------- STFT via WMMA
// Per block: one channel, 16 frames. Windowed frames staged in LDS, then
// GEMM (16 x 512) x (512 x 256) in 16x16 tiles with K-chunks of 4.
// Output layout S[f][t][k]: k<8 -> Re(ch k), k>=8 -> Im(ch k-8).
__global__ void __launch_bounds__(128) k_stft(const float* __restrict__ x,
                                              const float* __restrict__ Bc,
                                              const float* __restrict__ Bs,
                                              float* __restrict__ S) {
  int tt  = blockIdx.x;   // 0..62
  int c   = blockIdx.y;   // 0..7
  int tid = threadIdx.x;  // 128 threads = 4 waves

  __shared__ float win[NFFT];
  __shared__ float fr[16 * FROW];

  for (int i = tid; i < NFFT; i += 128) {
    float h = 0.5f - 0.5f * cosf((float)i * (TWO_PI / (float)NFFT));
    win[i] = sqrtf(h);                     // periodic sqrt-Hann
  }
  __syncthreads();

  int t0 = tt * 16;
  for (int e = tid; e < 16 * NFFT; e += 128) {
    int tl = e >> 9, n = e & (NFFT - 1);
    int t  = t0 + tl;
    float v = 0.0f;
    if (t < NT) {
      int idx = t * HOP - 256 + n;         // center=True, pad=256
      if (idx < 0) idx = -idx;             // reflect left
      else if (idx >= NSAMP) idx = 2 * NSAMP - 2 - idx;  // reflect right
      v = x[(size_t)idx * NCH + c] * win[n];
    }
    fr[tl * FROW + n] = v;
  }
  __syncthreads();

  int wave = tid >> 5, lane = tid & 31, lr = lane & 15, lh = lane >> 4;
  for (int ft = wave; ft < 16; ft += 4) {  // wave-uniform -> EXEC all 1s
    int f0 = ft * 16;
    v8f cre = {0, 0, 0, 0, 0, 0, 0, 0};
    v8f cim = {0, 0, 0, 0, 0, 0, 0, 0};
    for (int ch = 0; ch < 128; ++ch) {     // K = 512 in chunks of 4
      int kb = ch * 4 + lh * 2;            // A rows K = kb, kb+1
      int p  = ch * 2 + lh;                // pair index kb/2
      v2f a  = *(const v2f*)&fr[lr * FROW + kb];
      v2f bc = *(const v2f*)&Bc[((size_t)p * NF + f0 + lr) * 2];
      v2f bs = *(const v2f*)&Bs[((size_t)p * NF + f0 + lr) * 2];
      cre = wmma_f32(a, bc, cre);
      cim = wmma_f32(a, bs, cim);
    }
#pragma unroll
    for (int r = 0; r < 8; ++r) {
      int t = t0 + lh * 8 + r;             // D: M = 8*lh + r
      int f = f0 + lr;                     // D: N = lr
      float* dst = S + ((size_t)f * NTP + t) * 16;
      dst[c]     = cre[r];
      dst[8 + c] = cim[r];
    }
  }
}

// -------------------------------------------------- SRP einsum via WMMA
// Per block: one (freq, t-tile). 8 waves cover 23 angle tiles. K=16.
// Writes deterministic per-(f,ttile) partial sums of |resp| over 16 frames.
__global__ void __launch_bounds__(256) k_srp(const float* __restrict__ S,
                                             const float* __restrict__ Bre,
                                             const float* __restrict__ Bim,
                                             float* __restrict__ part) {
  int tt  = blockIdx.x;   // 0..62
  int f   = blockIdx.y;   // 0..255
  int tid = threadIdx.x;
  int wave = tid >> 5, lane = tid & 31, lr = lane & 15, lh = lane >> 4;
  int t0 = tt * 16;

  const float* Sf = S + ((size_t)f * NTP + t0) * 16;
  v2f afr[4];
#pragma unroll
  for (int ch = 0; ch < 4; ++ch) {
    int k = ch * 4 + lh * 2;
    afr[ch] = *(const v2f*)(Sf + lr * 16 + k);  // A[M=lr][k..k+1]
  }

  const float* Br = Bre + (size_t)f * 8 * NAP * 2;  // pair-interleaved slice
  const float* Bi = Bim + (size_t)f * 8 * NAP * 2;
  float* pout = part + ((size_t)f * NTT + tt) * NAP;

  for (int at = wave; at < NAT; at += 8) {      // wave-uniform
    int a0 = at * 16;
    v8f cre = {0, 0, 0, 0, 0, 0, 0, 0};
    v8f cim = {0, 0, 0, 0, 0, 0, 0, 0};
#pragma unroll
    for (int ch = 0; ch < 4; ++ch) {
      int p = ch * 2 + lh;                      // pair index k/2
      v2f br = *(const v2f*)&Br[((size_t)p * NAP + a0 + lr) * 2];
      v2f bi = *(const v2f*)&Bi[((size_t)p * NAP + a0 + lr) * 2];
      cre = wmma_f32(afr[ch], br, cre);
      cim = wmma_f32(afr[ch], bi, cim);
    }
    float s = 0.0f;
#pragma unroll
    for (int r = 0; r < 8; ++r)
      s += sqrtf(cre[r] * cre[r] + cim[r] * cim[r]);   // |resp| over 8 frames
    s += __shfl_xor(s, 16, 32);          // combine frames 0-7 with 8-15
    if (lh == 0) pout[a0 + lr] = s;      // angle = a0 + lr
  }
}

// --------------------------------------------- reduce partials over t-tiles
__global__ void __launch_bounds__(384) k_reduce(const float* __restrict__ part,
                                                float* __restrict__ acc) {
  int f = blockIdx.x;     // 0..255
  int a = threadIdx.x;    // 0..383
  if (a >= NAP) return;
  const float* p = part + (size_t)f * NTT * NAP + a;
  float s = 0.0f;
  for (int tt = 0; tt < NTT; ++tt) s += p[(size_t)tt * NAP];
  acc[(size_t)f * NAP + a] = s;
}

// ---------------------------- finalize: power map, mean over f, argmax, DOA
__global__ void __launch_bounds__(512) k_final(const float* __restrict__ acc,
                                               float* __restrict__ out) {
  int a = threadIdx.x;    // 0..511
  __shared__ float sv[512];
  __shared__ int   si[512];
  float pm = -1.0f;
  if (a < NA) {
    const float invT = 1.0f / (float)NT;
    float s = 0.0f;
    for (int f = 0; f < NF; ++f) {
      float m = acc[(size_t)f * NAP + a] * invT;  // mean_t |resp|
      s += m * m;                                 // power_map[f][a]
    }
    pm = s * (1.0f / (float)NF);                  // mean over f
    out[1 + a] = pm;
  }
  sv[a] = pm;
  si[a] = a;
  __syncthreads();
  for (int off = 256; off > 0; off >>= 1) {       // first-max argmax
    if (a < off) {
      float v2 = sv[a + off]; int i2 = si[a + off];
      if (v2 > sv[a] || (v2 == sv[a] && i2 < si[a])) { sv[a] = v2; si[a] = i2; }
    }
    __syncthreads();
  }
  if (a == 0) out[0] = (float)si[0] * 0.5f - 90.0f;
}

extern "C" void kernel_launch(void* const* d_in, const int* in_sizes, int n_in,
                              void* d_out, int out_size, void* d_ws,
                              size_t ws_size, hipStream_t stream) {
  (void)in_sizes; (void)n_in; (void)out_size; (void)ws_size;
  const float* x = (const float*)d_in[0];   // (256000, 8) float32
  float* out = (float*)d_out;               // [doa, power_map_mean(361)]

  // Carve workspace (total ~54 MB, L2-resident on MI455X's 192 MB L2).
  char* w = (char*)d_ws;
  auto al = [](size_t v) { return (v + 255) & ~(size_t)255; };
  size_t o = 0;
  float* Bc   = (float*)(w + o); o += al(sizeof(float) * (NFFT / 2) * NF * 2); // 0.5 MB
  float* Bs   = (float*)(w + o); o += al(sizeof(float) * (NFFT / 2) * NF * 2); // 0.5 MB
  float* S    = (float*)(w + o); o += al(sizeof(float) * NF * NTP * 16);       // 16.5 MB
  float* Bre  = (float*)(w + o); o += al(sizeof(float) * NF * 8 * NAP * 2);    // 6 MB
  float* Bim  = (float*)(w + o); o += al(sizeof(float) * NF * 8 * NAP * 2);    // 6 MB
  float* part = (float*)(w + o); o += al(sizeof(float) * NF * NTT * NAP);      // 23.7 MB
  float* acc  = (float*)(w + o); o += al(sizeof(float) * NF * NAP);            // 0.4 MB

  k_dft_init  <<<NFFT, NF, 0, stream>>>(Bc, Bs);
  k_steer_init<<<NF, 384, 0, stream>>>(Bre, Bim);
  k_stft      <<<dim3(NTT, NCH), 128, 0, stream>>>(x, Bc, Bs, S);
  k_srp       <<<dim3(NTT, NF), 256, 0, stream>>>(S, Bre, Bim, part);
  k_reduce    <<<NF, 384, 0, stream>>>(part, acc);
  k_final     <<<1, 512, 0, stream>>>(acc, out);
}